// GATConv_32487132627454
// MI455X (gfx1250) — compile-verified
//
#include <hip/hip_runtime.h>
#include <hip/hip_bf16.h>

// ---------------------------------------------------------------------------
// GATConv (heads=1) for MI455X / gfx1250, wave32.
//   h      = x @ W                      -> V_WMMA_F32_16X16X4_F32 (exact f32)
//   a_s/a_d= h @ att_src / att_dst      -> warp-per-node dot
//   a_e    = attr @ (W_edge @ att_edge) -> scalar per edge (linearity folds
//                                          the [*,11]@[11,128]@[128] chain)
//   loop a_e = scatter-mean of kept per-edge a_e (linearity again)
//   segment softmax over dst (atomic max / atomic add), then
//   out[dst] += alpha * h[src]          -> warp-per-edge, 4 ch/lane atomics
// ---------------------------------------------------------------------------

typedef __attribute__((ext_vector_type(2))) float v2f;
typedef __attribute__((ext_vector_type(8))) float v8f;

#define DINC 128
#define EDIMC 11
#define SLOPEC 0.2f
#define NEGF (-1e30f)

// ---------------- init: zero accumulators --------------------------------
__global__ void k_init(float* __restrict__ out, float* __restrict__ cnt,
                       float* __restrict__ sum_ae, float* __restrict__ denom,
                       int n, int total) {
  int i = blockIdx.x * blockDim.x + threadIdx.x;
  if (i < total) out[i] = 0.f;
  if (i < n) { cnt[i] = 0.f; sum_ae[i] = 0.f; denom[i] = 0.f; }
}

// ---------------- w_e = W_edge @ att_edge  (11-vector) -------------------
__global__ void k_we(const float* __restrict__ W_edge,
                     const float* __restrict__ att_edge,
                     float* __restrict__ w_e) {
  int j = threadIdx.x;
  if (j < EDIMC) {
    float s = 0.f;
    for (int c = 0; c < DINC; ++c) s += W_edge[j * DINC + c] * att_edge[c];
    w_e[j] = s;
  }
}

// ---------------- h = x @ W via WMMA f32 16x16x4 -------------------------
// One wave computes one 16x16 tile of h. 8 waves / block.
// A (16x4 f32):  lane<16 -> M=lane, K={k,k+1}; lane>=16 -> M=lane-16, K={k+2,k+3}
// B (4x16 f32):  v0 -> K=k+khalf, v1 -> K=k+khalf+1, N = lane&15
// D (16x16 f32): VGPR r -> M = 16*mt + r + 8*(lane>=16), N = 16*nt + (lane&15)
__global__ __launch_bounds__(256) void k_gemm_wmma(
    const float* __restrict__ x, const float* __restrict__ W,
    float* __restrict__ h, int mtiles) {
  const int wave = threadIdx.x >> 5;
  const int lane = threadIdx.x & 31;
  const int tile = blockIdx.x * 8 + wave;      // tiles = mtiles * 8 (C=128 -> 8 N-tiles)
  const int mt = tile >> 3;
  const int nt = tile & 7;
  if (mt >= mtiles) return;                    // wave-uniform: EXEC stays all-1s

  const int mrow  = mt * 16 + (lane & 15);
  const int ncol  = nt * 16 + (lane & 15);
  const int khalf = (lane >> 4) * 2;           // 0 or 2

  v8f acc = {};
  const float* __restrict__ arow = x + (size_t)mrow * DINC;
  #pragma unroll 4
  for (int k = 0; k < DINC; k += 4) {
    v2f a, b;
    a.x = arow[k + khalf];
    a.y = arow[k + khalf + 1];
    b.x = W[(size_t)(k + khalf)     * DINC + ncol];
    b.y = W[(size_t)(k + khalf + 1) * DINC + ncol];
    acc = __builtin_amdgcn_wmma_f32_16x16x4_f32(
        /*neg_a=*/false, a, /*neg_b=*/false, b,
        /*c_mod=*/(short)0, acc, /*reuse_a=*/false, /*reuse_b=*/false);
  }

  const int mbase = mt * 16 + ((lane >> 4) ? 8 : 0);
  #pragma unroll
  for (int r = 0; r < 8; ++r)
    h[(size_t)(mbase + r) * DINC + ncol] = acc[r];
}

// ---------------- a_s / a_d : warp-per-node dot --------------------------
__global__ void k_node_dots(const float* __restrict__ h,
                            const float* __restrict__ att_src,
                            const float* __restrict__ att_dst,
                            float* __restrict__ a_s, float* __restrict__ a_d,
                            int n) {
  int node = (blockIdx.x * blockDim.x + threadIdx.x) >> 5;
  int lane = threadIdx.x & 31;
  if (node >= n) return;
  const float* __restrict__ hp = h + (size_t)node * DINC;
  float ss = 0.f, sd = 0.f;
  #pragma unroll
  for (int c = lane; c < DINC; c += 32) {
    float v = hp[c];
    ss += v * att_src[c];
    sd += v * att_dst[c];
  }
  #pragma unroll
  for (int off = 16; off > 0; off >>= 1) {
    ss += __shfl_down(ss, off, 32);
    sd += __shfl_down(sd, off, 32);
  }
  if (lane == 0) { a_s[node] = ss; a_d[node] = sd; }
}

// ---------------- per-edge a_e + scatter cnt/sum for loop mean -----------
__global__ void k_edge_ae(const int* __restrict__ ei,
                          const float* __restrict__ edge_attr,
                          const float* __restrict__ w_e,
                          float* __restrict__ aeE,
                          float* __restrict__ cnt, float* __restrict__ sum_ae,
                          int E_) {
  int e = blockIdx.x * blockDim.x + threadIdx.x;
  if (e >= E_) return;
  int src = ei[e];
  int dst = ei[E_ + e];
  const float* __restrict__ ap = edge_attr + (size_t)e * EDIMC;
  float s = 0.f;
  #pragma unroll
  for (int j = 0; j < EDIMC; ++j) s += ap[j] * w_e[j];
  aeE[e] = s;
  if (src != dst) {
    atomicAdd(&cnt[dst], 1.0f);
    atomicAdd(&sum_ae[dst], s);
  }
}

// ---------------- self-loop logit; init segment-max ----------------------
__global__ void k_loop_logit(const float* __restrict__ a_s,
                             const float* __restrict__ a_d,
                             const float* __restrict__ cnt,
                             const float* __restrict__ sum_ae,
                             float* __restrict__ loopL, float* __restrict__ maxv,
                             int n) {
  int i = blockIdx.x * blockDim.x + threadIdx.x;
  if (i >= n) return;
  float ae = sum_ae[i] / fmaxf(cnt[i], 1.0f);   // scatter-mean (fill='mean')
  float z = a_s[i] + a_d[i] + ae;
  z = (z > 0.f) ? z : SLOPEC * z;
  loopL[i] = z;
  maxv[i] = z;   // self-loop always present -> valid init for segment max
}

// ---------------- atomic float max via sign-split int trick --------------
__device__ __forceinline__ void atomicMaxF(float* addr, float val) {
  if (val >= 0.f) atomicMax((int*)addr, __float_as_int(val));
  else            atomicMin((unsigned int*)addr, __float_as_uint(val));
}

// ---------------- edge logits + segment max ------------------------------
__global__ void k_edge_logit(const int* __restrict__ ei,
                             const float* __restrict__ a_s,
                             const float* __restrict__ a_d,
                             const float* __restrict__ aeE,
                             float* __restrict__ logitE, float* __restrict__ maxv,
                             int E_) {
  int e = blockIdx.x * blockDim.x + threadIdx.x;
  if (e >= E_) return;
  int src = ei[e];
  int dst = ei[E_ + e];
  if (src == dst) { logitE[e] = NEGF; return; }   // masked self-loop removal
  float z = a_s[src] + a_d[dst] + aeE[e];
  z = (z > 0.f) ? z : SLOPEC * z;
  logitE[e] = z;
  atomicMaxF(&maxv[dst], z);
}

// ---------------- p = exp(logit - m[dst]); denom = segsum(p) -------------
// Covers E edges + N self-loops; p overwrites the logit buffers in place.
__global__ void k_exp_denom(const int* __restrict__ ei,
                            float* __restrict__ logitE, float* __restrict__ loopL,
                            const float* __restrict__ maxv,
                            float* __restrict__ denom, int E_, int n) {
  int i = blockIdx.x * blockDim.x + threadIdx.x;
  if (i >= E_ + n) return;
  if (i < E_) {
    int dst = ei[E_ + i];
    float p = expf(logitE[i] - maxv[dst]);  // masked edge: exp(-1e30 - m) == 0
    logitE[i] = p;
    if (p != 0.f) atomicAdd(&denom[dst], p);
  } else {
    int nd = i - E_;
    float p = expf(loopL[nd] - maxv[nd]);
    loopL[nd] = p;
    atomicAdd(&denom[nd], p);
  }
}

// ---------------- out[dst] += alpha * h[src]  (warp per edge) ------------
__global__ void k_scatter(const int* __restrict__ ei,
                          const float* __restrict__ h,
                          const float* __restrict__ pE, const float* __restrict__ pL,
                          const float* __restrict__ denom,
                          float* __restrict__ out, int E_, int n) {
  int idx  = (blockIdx.x * blockDim.x + threadIdx.x) >> 5;  // warp-uniform
  int lane = threadIdx.x & 31;
  if (idx >= E_ + n) return;
  int src, dst; float p;
  if (idx < E_) { src = ei[idx]; dst = ei[E_ + idx]; p = pE[idx]; }
  else          { src = dst = idx - E_;              p = pL[idx - E_]; }
  if (p == 0.f) return;                    // masked edge contributes nothing
  float alpha = p / denom[dst];
  const float* __restrict__ hs = h + (size_t)src * DINC;
  float* __restrict__ od = out + (size_t)dst * DINC;
  int c = lane * 4;                        // 128 ch / 32 lanes = 4 per lane
  float4 hv = *reinterpret_cast<const float4*>(hs + c);
  atomicAdd(&od[c + 0], alpha * hv.x);
  atomicAdd(&od[c + 1], alpha * hv.y);
  atomicAdd(&od[c + 2], alpha * hv.z);
  atomicAdd(&od[c + 3], alpha * hv.w);
}

// ---------------------------------------------------------------------------
extern "C" void kernel_launch(void* const* d_in, const int* in_sizes, int n_in,
                              void* d_out, int out_size, void* d_ws, size_t ws_size,
                              hipStream_t stream) {
  const float* x        = (const float*)d_in[0];
  const int*   ei       = (const int*)  d_in[1];  // [2,E] int32 (jax default)
  const float* edgeAttr = (const float*)d_in[2];
  const float* W        = (const float*)d_in[3];
  const float* att_src  = (const float*)d_in[4];
  const float* att_dst  = (const float*)d_in[5];
  const float* W_edge   = (const float*)d_in[6];
  const float* att_edge = (const float*)d_in[7];
  float* out = (float*)d_out;

  const int N_ = in_sizes[0] / DINC;   // 50000
  const int E_ = in_sizes[1] / 2;      // 600000
  const int total = N_ * DINC;

  // workspace carve-up (floats)
  float* ws     = (float*)d_ws;
  float* h      = ws;  ws += (size_t)N_ * DINC;
  float* a_s    = ws;  ws += N_;
  float* a_d    = ws;  ws += N_;
  float* cnt    = ws;  ws += N_;
  float* sum_ae = ws;  ws += N_;
  float* maxv   = ws;  ws += N_;
  float* denom  = ws;  ws += N_;
  float* loopL  = ws;  ws += N_;
  float* aeE    = ws;  ws += E_;
  float* logitE = ws;  ws += E_;
  float* w_e    = ws;  ws += 16;

  const int B = 256;
  // 1) zero accumulators (out, cnt, sum_ae, denom)
  k_init<<<(total + B - 1) / B, B, 0, stream>>>(out, cnt, sum_ae, denom, N_, total);
  // 2) fold edge-attention weights: w_e = W_edge @ att_edge
  k_we<<<1, 32, 0, stream>>>(W_edge, att_edge, w_e);
  // 3) h = x @ W  (WMMA f32)
  const int mtiles = N_ / 16;                      // 3125 exact
  k_gemm_wmma<<<(mtiles * 8 + 7) / 8, B, 0, stream>>>(x, W, h, mtiles);
  // 4) a_s, a_d
  k_node_dots<<<((size_t)N_ * 32 + B - 1) / B, B, 0, stream>>>(h, att_src, att_dst, a_s, a_d, N_);
  // 5) per-edge a_e + loop-mean scatter
  k_edge_ae<<<(E_ + B - 1) / B, B, 0, stream>>>(ei, edgeAttr, w_e, aeE, cnt, sum_ae, E_);
  // 6) self-loop logit, init segment max
  k_loop_logit<<<(N_ + B - 1) / B, B, 0, stream>>>(a_s, a_d, cnt, sum_ae, loopL, maxv, N_);
  // 7) edge logits + segment max
  k_edge_logit<<<(E_ + B - 1) / B, B, 0, stream>>>(ei, a_s, a_d, aeE, logitE, maxv, E_);
  // 8) exp + segment sum (edges + self-loops)
  k_exp_denom<<<(E_ + N_ + B - 1) / B, B, 0, stream>>>(ei, logitE, loopL, maxv, denom, E_, N_);
  // 9) weighted scatter of h rows
  k_scatter<<<((size_t)(E_ + N_) * 32 + B - 1) / B, B, 0, stream>>>(
      ei, h, logitE, loopL, denom, out, E_, N_);
}